// BertSelfAttention_50551765074524
// MI455X (gfx1250) — compile-verified
//
#include <hip/hip_runtime.h>
#include <math.h>

typedef __attribute__((ext_vector_type(2))) float v2f;
typedef __attribute__((ext_vector_type(8))) float v8f;

#define N_CFG 4
#define B_CFG 8
#define T_CFG 2
#define E_CFG 1024
#define H_CFG 16
#define D_CFG 64
#define S_CFG 2048
#define L_CFG 1024
#define NB_CFG 32
#define CLEN 3074            // S + L + T
#define SROW 3088            // padded LDS row stride for score matrix

// -----------------------------------------------------------------------------
// Kernel 1: QKV projection.  out = hidden @ W^T + b  for W in {Wq,Wk,Wv}.
// Computed as out^T[n,m] = sum_k W[n,k]*hidden[m,k] with V_WMMA_F32_16X16X4_F32.
// One wave owns a 16-wide n-stripe of one matrix and all 64 m rows
// (4 accumulators), so each weight element is loaded exactly once.
// 192 wave-tasks total -> 24 blocks x 256 threads (8 waves).
// -----------------------------------------------------------------------------
__global__ __launch_bounds__(256) void qkv_kernel(
    const float* __restrict__ hid,
    const float* __restrict__ Wq, const float* __restrict__ bq,
    const float* __restrict__ Wk, const float* __restrict__ bk,
    const float* __restrict__ Wv, const float* __restrict__ bv,
    float* __restrict__ Qb, float* __restrict__ Kb, float* __restrict__ Vb)
{
    const int lane  = threadIdx.x & 31;
    const int w     = threadIdx.x >> 5;
    const int task  = blockIdx.x * 8 + w;        // 0..191
    const int mat   = task >> 6;                 // 0=Q, 1=K, 2=V
    const int nBase = (task & 63) * 16;

    const float* W    = (mat == 0) ? Wq : (mat == 1) ? Wk : Wv;
    const float* bias = (mat == 0) ? bq : (mat == 1) ? bk : bv;
    float*       Out  = (mat == 0) ? Qb : (mat == 1) ? Kb : Vb;

    const int lh = lane & 15;        // lane within half
    const int hh = lane >> 4;        // which half (K-offset 0 or 2)

    // A = W tile: row n = nBase+lh, K offset hh*2   (f32 16x4 A layout)
    const float* aPtr  = W   + (size_t)(nBase + lh) * E_CFG + hh * 2;
    // B = hidden^T tiles: col m = mBase+lh, K offset hh*2 (f32 4x16 B layout)
    const float* bPtr0 = hid + (size_t)( 0 + lh) * E_CFG + hh * 2;
    const float* bPtr1 = hid + (size_t)(16 + lh) * E_CFG + hh * 2;
    const float* bPtr2 = hid + (size_t)(32 + lh) * E_CFG + hh * 2;
    const float* bPtr3 = hid + (size_t)(48 + lh) * E_CFG + hh * 2;

    v8f c0, c1, c2, c3;
#pragma unroll
    for (int r = 0; r < 8; ++r) {                // C rows: n = nBase + r (+8 for hi half)
        float bb = bias[nBase + r + hh * 8];
        c0[r] = bb; c1[r] = bb; c2[r] = bb; c3[r] = bb;
    }

    for (int k0 = 0; k0 < E_CFG; k0 += 4) {
        __builtin_prefetch(aPtr + k0 + 64, 0, 0);
        v2f a  = *(const v2f*)(aPtr  + k0);
        v2f b0 = *(const v2f*)(bPtr0 + k0);
        v2f b1 = *(const v2f*)(bPtr1 + k0);
        v2f b2 = *(const v2f*)(bPtr2 + k0);
        v2f b3 = *(const v2f*)(bPtr3 + k0);
        c0 = __builtin_amdgcn_wmma_f32_16x16x4_f32(false, a, false, b0, (short)0, c0, false, false);
        c1 = __builtin_amdgcn_wmma_f32_16x16x4_f32(false, a, false, b1, (short)0, c1, false, false);
        c2 = __builtin_amdgcn_wmma_f32_16x16x4_f32(false, a, false, b2, (short)0, c2, false, false);
        c3 = __builtin_amdgcn_wmma_f32_16x16x4_f32(false, a, false, b3, (short)0, c3, false, false);
    }

    // D layout: VGPR r, lanes 0-15 -> (row n = nBase+r, col m = mBase+lane),
    //           lanes 16-31 -> (row n = nBase+8+r, col m = mBase+lane-16)
#pragma unroll
    for (int r = 0; r < 8; ++r) {
        int n = nBase + r + hh * 8;
        Out[(size_t)( 0 + lh) * E_CFG + n] = c0[r];
        Out[(size_t)(16 + lh) * E_CFG + n] = c1[r];
        Out[(size_t)(32 + lh) * E_CFG + n] = c2[r];
        Out[(size_t)(48 + lh) * E_CFG + n] = c3[r];
    }
}

// -----------------------------------------------------------------------------
// Kernel 2: attention.  One block per (n,h); 1024 threads = 32 waves.
// 16 query rows (8 beams x 2 tokens) batched so prefix K/V stream once per group.
// -----------------------------------------------------------------------------

// Compute a 16x16 score tile: D = Q(16x64) x Ktile^T(64x16) via 16 chained
// f32 16x16x4 WMMAs; scale + mask applied on the LDS write.
__device__ __forceinline__ void score_tile(
    const float* __restrict__ kTile,      // 16 rows x D, row stride D
    float (*sScore)[SROW],
    const float (*sQ)[68],
    const float* __restrict__ mask,
    int n, int col0, int beamOnly, int lane)
{
    const int lh = lane & 15;
    const int hh = lane >> 4;
    const float* bP = kTile + (size_t)lh * D_CFG + hh * 2;

    v8f c = {};
#pragma unroll
    for (int k0 = 0; k0 < D_CFG; k0 += 4) {
        v2f a;
        a.x = sQ[lh][k0 + hh * 2];
        a.y = sQ[lh][k0 + hh * 2 + 1];
        v2f b = *(const v2f*)(bP + k0);
        c = __builtin_amdgcn_wmma_f32_16x16x4_f32(false, a, false, b, (short)0, c, false, false);
    }
#pragma unroll
    for (int r = 0; r < 8; ++r) {
        int m = r + hh * 8;
        if (beamOnly >= 0 && (m >> 1) != beamOnly) continue;
        int nb  = n * B_CFG + (m >> 1);
        int col = col0 + lh;
        sScore[m][col] = c[r] * 0.125f + mask[(size_t)nb * CLEN + col];
    }
}

__global__ __launch_bounds__(1024) void attn_kernel(
    const float* __restrict__ mask,
    const float* __restrict__ pk, const float* __restrict__ pv,
    const float* __restrict__ ck, const float* __restrict__ cv,
    const float* __restrict__ Qb, const float* __restrict__ Kn,
    const float* __restrict__ Vn,
    float* __restrict__ out)
{
    __shared__ float sScore[16][SROW];     // full padded score rows (~193 KB)
    __shared__ float sQ[16][68];           // padded to kill bank conflicts
    __shared__ float sCtx[16][D_CFG];
    __shared__ float sInv[16];

    const int n    = blockIdx.x >> 4;
    const int h    = blockIdx.x & 15;
    const int tid  = threadIdx.x;
    const int lane = tid & 31;
    const int w    = tid >> 5;

    // ---- P0: load Q rows (m = beam*2 + t) into LDS, zero ctx accumulators
    {
        int m = tid >> 6, d = tid & 63;
        int nb = n * B_CFG + (m >> 1);
        sQ[m][d]   = Qb[((size_t)(nb * T_CFG + (m & 1))) * E_CFG + h * D_CFG + d];
        sCtx[m][d] = 0.0f;
    }
    __syncthreads();

    // ---- P1: prefix scores, 128 tiles over 32 waves
    {
        const float* Kp = pk + (size_t)(n * H_CFG + h) * S_CFG * D_CFG;
#pragma unroll
        for (int it = 0; it < 4; ++it) {
            int s0 = (w + it * 32) * 16;
            score_tile(Kp + (size_t)s0 * D_CFG, sScore, sQ, mask, n, s0, -1, lane);
        }
    }

    // ---- P2: per-beam current scores over past K, 512 tiles over 32 waves
#pragma unroll 1
    for (int i = 0; i < 16; ++i) {
        int task = w + i * 32;
        int b    = task >> 6;
        int s0   = (task & 63) * 16;
        int nb   = n * B_CFG + b;
        const float* Kc = ck + (((size_t)nb * H_CFG + h) * L_CFG + s0) * D_CFG;
        score_tile(Kc, sScore, sQ, mask, n, S_CFG + s0, b, lane);
    }

    // ---- P2b: the two new-key columns (k_new), 32 dot products on wave 31
    if (w == 31) {
        int b = lane >> 2, t = (lane >> 1) & 1, tp = lane & 1;
        int nb = n * B_CFG + b;
        const float* kn = Kn + ((size_t)(nb * T_CFG + tp)) * E_CFG + h * D_CFG;
        float acc = 0.0f;
#pragma unroll
        for (int d = 0; d < D_CFG; ++d) acc += sQ[b * 2 + t][d] * kn[d];
        int col = S_CFG + L_CFG + tp;
        sScore[b * 2 + t][col] = acc * 0.125f + mask[(size_t)nb * CLEN + col];
    }
    __syncthreads();

    // ---- P3: softmax, one wave per row; store unnormalized exp, keep 1/sum
    if (w < 16) {
        int m = w;
        float mx = -1e30f;
        for (int i = lane; i < CLEN; i += 32) mx = fmaxf(mx, sScore[m][i]);
#pragma unroll
        for (int off = 16; off; off >>= 1) mx = fmaxf(mx, __shfl_xor(mx, off, 32));
        float sum = 0.0f;
        for (int i = lane; i < CLEN; i += 32) {
            float e = __expf(sScore[m][i] - mx);
            sScore[m][i] = e;
            sum += e;
        }
#pragma unroll
        for (int off = 16; off; off >>= 1) sum += __shfl_xor(sum, off, 32);
        if (lane == 0) sInv[m] = 1.0f / sum;
    }
    __syncthreads();

    // ---- P4: prefix P*V — each wave streams a 64-row V chunk (coalesced
    //          float2 loads), prob broadcasts from LDS, ds_add_f32 accumulate
    {
        const float* Vp = pv + (size_t)(n * H_CFG + h) * S_CFG * D_CFG
                             + (size_t)w * 64 * D_CFG;
        const int d0 = lane * 2;
        float ax[16], ay[16];
#pragma unroll
        for (int m = 0; m < 16; ++m) { ax[m] = 0.0f; ay[m] = 0.0f; }
        for (int s = 0; s < 64; ++s) {
            const float* vr = Vp + (size_t)s * D_CFG + d0;
            __builtin_prefetch(vr + 8 * D_CFG, 0, 0);
            float vx = vr[0], vy = vr[1];
            int sg = w * 64 + s;
#pragma unroll
            for (int m = 0; m < 16; ++m) {
                float p = sScore[m][sg];           // LDS broadcast
                ax[m] += p * vx; ay[m] += p * vy;
            }
        }
#pragma unroll
        for (int m = 0; m < 16; ++m) {
            atomicAdd(&sCtx[m][d0],     ax[m]);
            atomicAdd(&sCtx[m][d0 + 1], ay[m]);
        }
    }

    // ---- P5: per-beam current P*V (past V + 2 new V rows); wave -> (beam, chunk)
    {
        int b = w >> 2, chunk = w & 3;
        int nb = n * B_CFG + b;
        const int d0 = lane * 2;
        float ax0 = 0, ay0 = 0, ax1 = 0, ay1 = 0;
        const float* Vc = cv + ((size_t)nb * H_CFG + h) * L_CFG * D_CFG;
        int sBeg = chunk * 256;
        for (int s = sBeg; s < sBeg + 256; ++s) {
            const float* vr = Vc + (size_t)s * D_CFG + d0;
            __builtin_prefetch(vr + 8 * D_CFG, 0, 0);
            float vx = vr[0], vy = vr[1];
            float p0 = sScore[b * 2 + 0][S_CFG + s];
            float p1 = sScore[b * 2 + 1][S_CFG + s];
            ax0 += p0 * vx; ay0 += p0 * vy;
            ax1 += p1 * vx; ay1 += p1 * vy;
        }
        if (chunk == 3) {
#pragma unroll
            for (int tp = 0; tp < 2; ++tp) {
                const float* vr = Vn + ((size_t)(nb * T_CFG + tp)) * E_CFG
                                     + h * D_CFG + d0;
                float vx = vr[0], vy = vr[1];
                float p0 = sScore[b * 2 + 0][S_CFG + L_CFG + tp];
                float p1 = sScore[b * 2 + 1][S_CFG + L_CFG + tp];
                ax0 += p0 * vx; ay0 += p0 * vy;
                ax1 += p1 * vx; ay1 += p1 * vy;
            }
        }
        atomicAdd(&sCtx[b * 2 + 0][d0],     ax0);
        atomicAdd(&sCtx[b * 2 + 0][d0 + 1], ay0);
        atomicAdd(&sCtx[b * 2 + 1][d0],     ax1);
        atomicAdd(&sCtx[b * 2 + 1][d0 + 1], ay1);
    }
    __syncthreads();

    // ---- P6: normalize and write ctx -> out[nb, t, h*D + d]
    {
        int m = tid >> 6, d = tid & 63;
        int nb = n * B_CFG + (m >> 1);
        out[((size_t)(nb * T_CFG + (m & 1))) * E_CFG + h * D_CFG + d] =
            sCtx[m][d] * sInv[m];
    }
}

// -----------------------------------------------------------------------------
extern "C" void kernel_launch(void* const* d_in, const int* in_sizes, int n_in,
                              void* d_out, int out_size, void* d_ws, size_t ws_size,
                              hipStream_t stream)
{
    const float* hid  = (const float*)d_in[0];
    const float* mask = (const float*)d_in[1];
    const float* pk   = (const float*)d_in[2];
    const float* pv   = (const float*)d_in[3];
    const float* ck   = (const float*)d_in[4];
    const float* cv   = (const float*)d_in[5];
    const float* Wq   = (const float*)d_in[6];
    const float* bq   = (const float*)d_in[7];
    const float* Wk   = (const float*)d_in[8];
    const float* bk   = (const float*)d_in[9];
    const float* Wv   = (const float*)d_in[10];
    const float* bv   = (const float*)d_in[11];
    float* out = (float*)d_out;

    float* Qb = (float*)d_ws;                    // 64 x 1024
    float* Kb = Qb + (size_t)NB_CFG * T_CFG * E_CFG;
    float* Vb = Kb + (size_t)NB_CFG * T_CFG * E_CFG;

    qkv_kernel<<<24, 256, 0, stream>>>(hid, Wq, bq, Wk, bk, Wv, bv, Qb, Kb, Vb);
    attn_kernel<<<N_CFG * H_CFG, 1024, 0, stream>>>(mask, pk, pv, ck, cv,
                                                    Qb, Kb, Vb, out);
}